// AnchorFreeGenerator_15006615733625
// MI455X (gfx1250) — compile-verified
//
#include <hip/hip_runtime.h>

#define N_GT 256
#define BLK  256
#define PPT  2   // points per thread

typedef int v4i __attribute__((vector_size(16)));
typedef __attribute__((address_space(1))) v4i* gas1_v4i_ptr;   // global
typedef __attribute__((address_space(3))) v4i* gas3_v4i_ptr;   // LDS

__global__ __launch_bounds__(BLK) void centerness_max_kernel(
    const float* __restrict__ points,
    const float* __restrict__ gt,      // [N_GT][4] = x0,y0,x1,y1
    float* __restrict__ out,
    int n_points)
{
    __shared__ __align__(16) float4 sbox[N_GT];   // 4 KB box table

    const int tid = threadIdx.x;

    // ---- stage all 256 boxes into LDS: 16 bytes per thread, async copy ----
    {
        const float* g = gt + tid * 4;
#if __has_builtin(__builtin_amdgcn_global_load_async_to_lds_b128)
        __builtin_amdgcn_global_load_async_to_lds_b128(
            (gas1_v4i_ptr)g,
            (gas3_v4i_ptr)(sbox + tid),
            /*offset=*/0, /*cpol=*/0);
#if __has_builtin(__builtin_amdgcn_s_wait_asynccnt)
        __builtin_amdgcn_s_wait_asynccnt(0);
#else
        asm volatile("s_wait_asynccnt 0" ::: "memory");
#endif
#else
        // Fallback: plain global->VGPR->LDS copy
        sbox[tid] = *(const float4*)g;
#endif
    }
    __syncthreads();

    const int base = blockIdx.x * (BLK * PPT) + tid;

    float px[PPT], py[PPT], m[PPT];
    bool  ok[PPT];
#pragma unroll
    for (int p = 0; p < PPT; ++p) {
        const int idx = base + p * BLK;
        ok[p] = (idx < n_points);
        float2 pt = ok[p] ? *(const float2*)(points + 2 * (size_t)idx)
                          : make_float2(0.0f, 0.0f);
        px[p] = pt.x;
        py[p] = pt.y;
        m[p]  = 0.0f;
    }

    const float eps = 1e-6f;

#pragma unroll 4
    for (int j = 0; j < N_GT; ++j) {
        const float4 b = sbox[j];   // ds_load_b128, same addr across wave -> broadcast
#pragma unroll
        for (int p = 0; p < PPT; ++p) {
            const float l  = fmaxf(px[p] - b.x, 0.0f);
            const float t  = fmaxf(py[p] - b.y, 0.0f);
            const float r  = fmaxf(b.z - px[p], 0.0f);
            const float bt = fmaxf(b.w - py[p], 0.0f);
            // sqrt(l*r/max(l+r,eps)) * sqrt(t*b/max(t+b,eps))
            //   == sqrt( num / den )            with num = l*r*t*b >= 0
            //   == num * rsqrt(num * den)       (clamped so num==0 -> 0, not NaN)
            const float num  = (l * r) * (t * bt);
            const float den  = fmaxf(l + r, eps) * fmaxf(t + bt, eps);
            const float rad  = fmaxf(num * den, 1e-30f);
            const float c    = num * __builtin_amdgcn_rsqf(rad);
            m[p] = fmaxf(m[p], c);
        }
    }

#pragma unroll
    for (int p = 0; p < PPT; ++p) {
        const int idx = base + p * BLK;
        if (ok[p]) out[idx] = m[p];
    }
}

extern "C" void kernel_launch(void* const* d_in, const int* in_sizes, int n_in,
                              void* d_out, int out_size, void* d_ws, size_t ws_size,
                              hipStream_t stream) {
    const float* points = (const float*)d_in[0];   // [N,2] f32
    const float* gt     = (const float*)d_in[1];   // [256,4] f32
    // d_in[2] = strides: unused by the reference math
    float* out = (float*)d_out;                    // [N] f32

    const int n_points = in_sizes[0] / 2;
    const int grid = (n_points + BLK * PPT - 1) / (BLK * PPT);
    centerness_max_kernel<<<grid, BLK, 0, stream>>>(points, gt, out, n_points);
}